// SimpleCRFHead_27625229648162
// MI455X (gfx1250) — compile-verified
//
#include <hip/hip_runtime.h>
#include <hip/hip_bf16.h>

// SimpleCRFHead on gfx1250 (MI455X).
// - Exact fwd/bwd split: alpha scan t=0..127 and beta scan t=255..127 run as
//   independent waves (1024 waves total), combined via logsumexp(alpha+beta).
// - Each wave carries 16 sequences in "CD layout": state j = VGPR row, seq n =
//   lane (j=8 rides in the upper half-lanes of row 0). The per-step 9x9
//   logsumexp matvec runs in the exp2 domain as D = A x B on the matrix pipe:
//   A = exp(tran) (constant, zero-padded 16x12, 3 tiles), B = per-seq softmax
//   vectors with sequences as columns -> D lands back in CD layout, so there
//   is NO per-step transpose. 3x V_WMMA_F32_16X16X4_F32 per step.
// - inputs/path staged to LDS with double-buffered global_load_async_to_lds_*
//   (ASYNCcnt), which also makes the x[path_t] / tran[p,p'] gathers 1 ds_load.

#define NST    9
#define LSEQ   256
#define BATCH  8192
#define CT     8               // timesteps per staged chunk
#define NCHUNK 16              // 128 steps per direction
#define L2E    1.44269504088896340736f
#define LN2    0.69314718055994530942f

// LDS layout (dwords). x rows: 10 dw (9 payload + pad) -> conflict-free reads.
#define TRAN_DW   96
#define XBUF_DW   (CT * 16 * 10)        // 1280 dw per x buffer
#define PBUF_DW   (16 * 10)             // 160 dw per path buffer
#define WAVE_DW   (2 * XBUF_DW + 2 * PBUF_DW)   // 2880 dw per wave
#define NWAVES    2
#define BLOCK_T   64
#define SMEM_BYTES (((TRAN_DW + NWAVES * WAVE_DW) * 4) + 64)  // +pad for benign OOB-in-LDS reads

typedef float v2f __attribute__((ext_vector_type(2)));
typedef float v8f __attribute__((ext_vector_type(8)));

#if __has_builtin(__builtin_amdgcn_exp2f)
#define EXP2F(x) __builtin_amdgcn_exp2f(x)
#else
#define EXP2F(x) exp2f(x)
#endif
#if __has_builtin(__builtin_amdgcn_logf)
#define LOG2F(x) __builtin_amdgcn_logf(x)   // hardware v_log_f32 is log2
#else
#define LOG2F(x) log2f(x)
#endif

#define WMMA4(d, av, bv)                                                       \
  d = __builtin_amdgcn_wmma_f32_16x16x4_f32(false, av, false, bv, (short)0, d, \
                                            false, false)

#define ASYNC_B128(ldsaddr, gaddr)                                             \
  asm volatile("global_load_async_to_lds_b128 %0, %1, off offset:0"            \
               :: "v"(ldsaddr), "v"(gaddr) : "memory")
#define ASYNC_B32(ldsaddr, gaddr)                                              \
  asm volatile("global_load_async_to_lds_b32 %0, %1, off offset:0"             \
               :: "v"(ldsaddr), "v"(gaddr) : "memory")
#define WAIT_ASYNC(N) asm volatile("s_wait_asynccnt " #N ::: "memory")

__global__ __launch_bounds__(BLOCK_T) void crf_scan_kernel(
    const float* __restrict__ inputs, const int* __restrict__ path,
    const float* __restrict__ tran, const float* __restrict__ init,
    float* __restrict__ ws)
{
  extern __shared__ float sm[];
  int* smi = (int*)sm;
  const int tid  = threadIdx.x;
  const int lane = tid & 31;
  const int wid  = tid >> 5;
  const int nn   = lane & 15;          // sequence slot within wave
  const bool lo  = lane < 16;

  // Stage tran (81) + init (9) into LDS once per block.
  for (int i = tid; i < 90; i += BLOCK_T)
    sm[i] = (i < 81) ? tran[i] : init[i - 81];
  __syncthreads();

  const int wave_global = blockIdx.x * NWAVES + wid;   // 0..1023
  const int is_bwd = wave_global & 1;
  const int b = (wave_global >> 1) * 16 + nn;          // sequence index

  // --- Constant A operand: 16x12 (3 tiles of 16x4), zero-padded exp(tran). ---
  // fwd: D[j][n] = sum_k expT[k][j] * e_k[n]  -> A[m][k] = expT[k][m]
  // bwd: D[i][n] = sum_j expT[i][j] * g_j[n]  -> A[m][k] = expT[m][k]
  v2f A[3];
  {
    const int mr = nn;
    #pragma unroll
    for (int w = 0; w < 3; ++w) {
      const int k0 = 4 * w + (lo ? 0 : 2);
      const int k1 = 4 * w + (lo ? 1 : 3);
      float v0 = 0.f, v1 = 0.f;
      if (mr < NST) {
        if (k0 < NST) {
          const int id0 = is_bwd ? (mr * 9 + k0) : (k0 * 9 + mr);
          v0 = EXP2F(sm[id0] * L2E);
        }
        if (k1 < NST) {
          const int id1 = is_bwd ? (mr * 9 + k1) : (k1 * 9 + mr);
          v1 = EXP2F(sm[id1] * L2E);
        }
      }
      A[w] = v2f{v0, v1};
    }
  }

  // --- Async staging setup ---
  const int xdw = TRAN_DW + wid * WAVE_DW;        // dword base of x buffers
  const int pdw = xdw + 2 * XBUF_DW;              // dword base of path buffers
  const int half16 = lo ? 0 : 16;
  const unsigned long long grow =
      (unsigned long long)inputs + (unsigned long long)b * (LSEQ * NST * 4);
  const unsigned long long prowg =
      (unsigned long long)path + (unsigned long long)b * (LSEQ * 4);
  const unsigned lx0 = (unsigned)(xdw * 4 + nn * 40);
  const unsigned lp0 = (unsigned)(pdw * 4 + nn * 40);

  auto issue_chunk = [&](int c) {
    const int t0  = is_bwd ? (248 - CT * c) : (CT * c);
    const int buf = c & 1;
    const unsigned lxb = lx0 + (unsigned)buf * (XBUF_DW * 4);
    const unsigned long long gxb = grow + (unsigned long long)(t0 * 36);
    #pragma unroll
    for (int r = 0; r < CT; ++r) {
      const unsigned lA = lxb + (unsigned)(r * 640) + (unsigned)half16;
      const unsigned lB = lxb + (unsigned)(r * 640) + 32u;
      const unsigned long long gA = gxb + (unsigned long long)(r * 36 + half16);
      const unsigned long long gB = gxb + (unsigned long long)(r * 36 + 32);
      ASYNC_B128(lA, gA);   // lanes 0-15: bytes 0-15, lanes 16-31: bytes 16-31
      ASYNC_B32(lB, gB);    // byte 32-35 (both halves duplicate, same data)
    }
    const unsigned lP = lp0 + (unsigned)buf * (PBUF_DW * 4) + (unsigned)half16;
    const unsigned long long gP = prowg + (unsigned long long)(t0 * 4 + half16);
    ASYNC_B128(lP, gP);     // 8 path ints per row, halves copy 16B each
  };

  float a[8];        // CD-layout state: row r = state r (upper row0 = state 8)
  float ps = 0.f;

  issue_chunk(0);

  if (!is_bwd) {
    int pprev = 0;
    for (int c = 0; c < NCHUNK; ++c) {
      if (c + 1 < NCHUNK) { issue_chunk(c + 1); WAIT_ASYNC(17); }
      else                { WAIT_ASYNC(0); }
      const int buf = c & 1;
      #pragma unroll
      for (int r = 0; r < CT; ++r) {
        const float* xr = sm + xdw + buf * XBUF_DW + r * 160 + nn * 10 + (lo ? 0 : 8);
        float x[8];
        #pragma unroll
        for (int k = 0; k < 8; ++k) x[k] = xr[k];
        const int p = smi[pdw + buf * PBUF_DW + nn * 10 + r];
        const float emit = xr[p];                    // valid in lower lanes
        if (c == 0 && r == 0) {
          ps = emit + sm[81 + p];
          #pragma unroll
          for (int j = 0; j < 8; ++j)
            a[j] = (sm[81 + j + (lo ? 0 : 8)] + x[j]) * L2E;
        } else {
          ps += emit + sm[pprev * 9 + p];
          float mlo = a[0];
          #pragma unroll
          for (int j = 1; j < 8; ++j) mlo = fmaxf(mlo, a[j]);
          mlo = fmaxf(mlo, __shfl_xor(a[0], 16, 32));    // fold in state 8
          const float msw = __shfl_xor(mlo, 16, 32);
          const float m = lo ? mlo : msw;
          float e[8];
          #pragma unroll
          for (int j = 0; j < 8; ++j) e[j] = EXP2F(a[j] - m);
          const float s0 = __shfl_xor(e[0], 16, 32);
          const float s2 = __shfl_xor(e[2], 16, 32);
          const float s3 = __shfl_xor(e[3], 16, 32);
          const float s6 = __shfl_xor(e[6], 16, 32);
          const float s7 = __shfl_xor(e[7], 16, 32);
          const v2f b0 = v2f{lo ? e[0] : s2, lo ? e[1] : s3};   // rows k=0..3
          const v2f b1 = v2f{lo ? e[4] : s6, lo ? e[5] : s7};   // rows k=4..7
          const v2f b2 = v2f{lo ? s0 : 0.f, 0.f};               // rows k=8..11
          v8f d = {};
          WMMA4(d, A[0], b0);
          WMMA4(d, A[1], b1);
          WMMA4(d, A[2], b2);
          #pragma unroll
          for (int j = 0; j < 8; ++j)
            a[j] = fmaf(x[j], L2E, m + LOG2F(d[j]));
        }
        pprev = p;
      }
    }
    const float a8 = __shfl_xor(a[0], 16, 32);
    if (lo) {
      float* w = ws + (size_t)b * 20;
      #pragma unroll
      for (int j = 0; j < 8; ++j) w[j] = a[j];
      w[8] = a8;
      w[9] = ps;
    }
  } else {
    #pragma unroll
    for (int j = 0; j < 8; ++j) a[j] = 0.f;          // beta'_255 = 0
    int pnext = 0;
    for (int c = 0; c < NCHUNK; ++c) {
      if (c + 1 < NCHUNK) { issue_chunk(c + 1); WAIT_ASYNC(17); }
      else                { WAIT_ASYNC(0); }
      const int buf = c & 1;
      #pragma unroll
      for (int r = CT - 1; r >= 0; --r) {            // t descends: t = t0 + r
        const float* xr = sm + xdw + buf * XBUF_DW + r * 160 + nn * 10 + (lo ? 0 : 8);
        float x[8];
        #pragma unroll
        for (int k = 0; k < 8; ++k) x[k] = xr[k];
        const int p = smi[pdw + buf * PBUF_DW + nn * 10 + r];
        ps += xr[p];
        if (!(c == 0 && r == CT - 1)) ps += sm[p * 9 + pnext];
        float h[8];
        #pragma unroll
        for (int j = 0; j < 8; ++j) h[j] = fmaf(x[j], L2E, a[j]);
        float mlo = h[0];
        #pragma unroll
        for (int j = 1; j < 8; ++j) mlo = fmaxf(mlo, h[j]);
        mlo = fmaxf(mlo, __shfl_xor(h[0], 16, 32));
        const float msw = __shfl_xor(mlo, 16, 32);
        const float m = lo ? mlo : msw;
        float g[8];
        #pragma unroll
        for (int j = 0; j < 8; ++j) g[j] = EXP2F(h[j] - m);
        const float s0 = __shfl_xor(g[0], 16, 32);
        const float s2 = __shfl_xor(g[2], 16, 32);
        const float s3 = __shfl_xor(g[3], 16, 32);
        const float s6 = __shfl_xor(g[6], 16, 32);
        const float s7 = __shfl_xor(g[7], 16, 32);
        const v2f b0 = v2f{lo ? g[0] : s2, lo ? g[1] : s3};
        const v2f b1 = v2f{lo ? g[4] : s6, lo ? g[5] : s7};
        const v2f b2 = v2f{lo ? s0 : 0.f, 0.f};
        v8f d = {};
        WMMA4(d, A[0], b0);
        WMMA4(d, A[1], b1);
        WMMA4(d, A[2], b2);
        #pragma unroll
        for (int j = 0; j < 8; ++j) a[j] = m + LOG2F(d[j]);
        pnext = p;
      }
    }
    const int p127 = path[(size_t)b * LSEQ + 127];
    ps += sm[p127 * 9 + pnext];                      // boundary pair (p127,p128)
    const float a8 = __shfl_xor(a[0], 16, 32);
    if (lo) {
      float* w = ws + (size_t)b * 20 + 10;
      #pragma unroll
      for (int j = 0; j < 8; ++j) w[j] = a[j];
      w[8] = a8;
      w[9] = ps;
    }
  }
}

__global__ void crf_combine_kernel(const float* __restrict__ ws,
                                   float* __restrict__ out)
{
  const int i = blockIdx.x * blockDim.x + threadIdx.x;
  if (i >= BATCH) return;
  const float* w = ws + (size_t)i * 20;
  float v[NST];
  #pragma unroll
  for (int j = 0; j < NST; ++j) v[j] = w[j] + w[10 + j];  // alpha' + beta'
  float m = v[0];
  #pragma unroll
  for (int j = 1; j < NST; ++j) m = fmaxf(m, v[j]);
  float s = 0.f;
  #pragma unroll
  for (int j = 0; j < NST; ++j) s += EXP2F(v[j] - m);
  const float score = (m + LOG2F(s)) * LN2;               // back to nats
  out[i] = score - (w[9] + w[19]);
}

extern "C" void kernel_launch(void* const* d_in, const int* in_sizes, int n_in,
                              void* d_out, int out_size, void* d_ws, size_t ws_size,
                              hipStream_t stream) {
  (void)in_sizes; (void)n_in; (void)out_size; (void)ws_size;
  const float* inputs = (const float*)d_in[0];
  const int*   path   = (const int*)d_in[1];
  const float* tran   = (const float*)d_in[2];
  const float* init   = (const float*)d_in[3];
  float* ws  = (float*)d_ws;     // needs 8192*20*4 = 640 KB
  float* out = (float*)d_out;

  // 1024 waves: 512 blocks x 2 waves (fwd + bwd halves, 16 sequences each).
  crf_scan_kernel<<<512, BLOCK_T, SMEM_BYTES, stream>>>(inputs, path, tran, init, ws);
  crf_combine_kernel<<<32, 256, 0, stream>>>(ws, out);
}